// Density_loss_83932250898497
// MI455X (gfx1250) — compile-verified
//
#include <hip/hip_runtime.h>

typedef __attribute__((ext_vector_type(2))) float v2f;
typedef __attribute__((ext_vector_type(8))) float v8f;

#define KNN_K 16

constexpr int Bsz  = 8;
constexpr int Npts = 2048;
constexpr int TILE = 16;
constexpr int NRT  = Npts / TILE;       // 128 row tiles per batch
constexpr int NJ   = Npts / (2 * TILE); // 64 iterations, 2 column tiles each

// One wave (32 threads) per 16-row tile. Per iteration: two fp32 WMMAs produce
// a 16x32 squared-distance block (D = A*B + C with A=(-2x,-2y,-2z,1),
// B=(x,y,z,|x|^2), C=|x_i|^2). All 32 lanes then scan half a row each,
// maintaining a sorted 16-smallest list via a min/max carry chain; the two
// half-row lists are merged at the end (two-pointer over sorted lists).
__global__ __launch_bounds__(32) void knn_rowmean_kernel(const float* __restrict__ x,
                                                         float* __restrict__ rowmean)
{
    __shared__ float rowsq[TILE];
    __shared__ float tileD[TILE * 32];      // 16 rows x 32 cols staging
    __shared__ float mergebuf[32 * KNN_K];  // per-lane sorted lists

    const int  lane = threadIdx.x;   // 0..31
    const bool hi   = lane >= 16;
    const int  l15  = lane & 15;

    const int blk = blockIdx.x;      // b * NRT + itile
    const int b   = blk / NRT;
    const int i0  = (blk % NRT) * TILE;

    // ---- A fragment: A[i,:] = (-2x, -2y, -2z, 1) ----
    // 16x4 f32 A layout: lanes 0-15 hold K=0,1 ; lanes 16-31 hold K=2,3.
    const float* pi = x + ((size_t)(b * Npts + i0 + l15)) * 3;
    float ax = pi[0], ay = pi[1], az = pi[2];
    float asq = ax * ax + ay * ay + az * az;
    v2f afrag;
    afrag.x = hi ? (-2.0f * az) : (-2.0f * ax);
    afrag.y = hi ? 1.0f         : (-2.0f * ay);

    if (lane < TILE) rowsq[lane] = asq;
    __syncthreads();

    // ---- C accumulator: C[i][j] = |x_i|^2 ----
    v8f cfrag;
    #pragma unroll
    for (int r = 0; r < 8; ++r) cfrag[r] = rowsq[r + (hi ? 8 : 0)];

    // ---- per-lane sorted 16-smallest list (ascending; list[15] = worst) ----
    float list[KNN_K];
    #pragma unroll
    for (int t = 0; t < KNN_K; ++t) list[t] = 3.4e38f;

    const int rbase = hi ? 8 : 0;

    for (int jt = 0; jt < NJ; ++jt) {
        const int j0 = jt * 32;
        // ---- B fragments for column tiles j0..j0+15 and j0+16..j0+31 ----
        const float* pj0 = x + ((size_t)(b * Npts + j0 + l15)) * 3;
        const float* pj1 = pj0 + (size_t)TILE * 3;
        float c0x = pj0[0], c0y = pj0[1], c0z = pj0[2];
        float c1x = pj1[0], c1y = pj1[1], c1z = pj1[2];
        float c0sq = c0x * c0x + c0y * c0y + c0z * c0z;
        float c1sq = c1x * c1x + c1y * c1y + c1z * c1z;
        v2f b0, b1;
        b0.x = hi ? c0z  : c0x;  b0.y = hi ? c0sq : c0y;
        b1.x = hi ? c1z  : c1x;  b1.y = hi ? c1sq : c1y;

        // Two distance tiles per iteration (uniform EXEC at WMMA).
        v8f d0 = __builtin_amdgcn_wmma_f32_16x16x4_f32(
            false, afrag, false, b0, (short)0, cfrag, false, false);
        v8f d1 = __builtin_amdgcn_wmma_f32_16x16x4_f32(
            false, afrag, false, b1, (short)0, cfrag, false, false);

        #pragma unroll
        for (int r = 0; r < 8; ++r) {
            tileD[(r + rbase) * 32 + l15]      = d0[r];
            tileD[(r + rbase) * 32 + 16 + l15] = d1[r];
        }
        __syncthreads();

        // Every lane scans 16 candidates: row l15, cols (hi?16:0)+0..15.
        const float4* rowp = (const float4*)&tileD[l15 * 32 + (hi ? 16 : 0)];
        #pragma unroll
        for (int q = 0; q < 4; ++q) {
            float4 v4 = rowp[q];
            float vs[4] = { v4.x, v4.y, v4.z, v4.w };
            #pragma unroll
            for (int e = 0; e < 4; ++e) {
                float v = vs[e];
                if (v < list[KNN_K - 1]) {
                    // sorted insert: keep smaller at each slot, carry larger up
                    #pragma unroll
                    for (int t = 0; t < KNN_K; ++t) {
                        float lo = fminf(list[t], v);
                        v        = fmaxf(list[t], v);
                        list[t]  = lo;
                    }
                }
            }
        }
        __syncthreads();
    }

    // ---- merge the two sorted half-row lists (lane L with lane L+16) ----
    #pragma unroll
    for (int t = 0; t < KNN_K; ++t) mergebuf[lane * KNN_K + t] = list[t];
    __syncthreads();

    if (lane < TILE) {
        const float* La = &mergebuf[lane * KNN_K];
        const float* Lb = &mergebuf[(lane + 16) * KNN_K];
        int ia = 0, ib = 0;
        float s = 0.0f;
        #pragma unroll
        for (int t = 0; t < KNN_K; ++t) {
            float va = La[ia];
            float vb = Lb[ib];
            bool ta = (va <= vb);
            s  += ta ? va : vb;
            ia += ta ? 1 : 0;
            ib += ta ? 0 : 1;
        }
        rowmean[(size_t)b * Npts + i0 + lane] = s * (1.0f / KNN_K);
    }
}

// rm holds [2][B][N] row means: seed at sb=0..7, gt at sb=8..15.
// loss = mean_b( (mean_n seed_rm - mean_n gt_rm)^2 )
__global__ __launch_bounds__(256) void density_loss_reduce(const float* __restrict__ rm,
                                                           float* __restrict__ out)
{
    __shared__ float red[256];
    __shared__ float means[2 * Bsz];
    const int tid = threadIdx.x;

    for (int sb = 0; sb < 2 * Bsz; ++sb) {
        float acc = 0.0f;
        for (int n = tid; n < Npts; n += 256) acc += rm[(size_t)sb * Npts + n];
        red[tid] = acc;
        __syncthreads();
        for (int off = 128; off > 0; off >>= 1) {
            if (tid < off) red[tid] += red[tid + off];
            __syncthreads();
        }
        if (tid == 0) means[sb] = red[0] * (1.0f / Npts);
        __syncthreads();
    }

    if (tid == 0) {
        float loss = 0.0f;
        for (int b = 0; b < Bsz; ++b) {
            float dd = means[b] - means[Bsz + b];
            loss += dd * dd;
        }
        out[0] = loss * (1.0f / Bsz);
    }
}

extern "C" void kernel_launch(void* const* d_in, const int* in_sizes, int n_in,
                              void* d_out, int out_size, void* d_ws, size_t ws_size,
                              hipStream_t stream)
{
    (void)in_sizes; (void)n_in; (void)out_size; (void)ws_size;

    const float* seed = (const float*)d_in[0];
    const float* gts  = (const float*)d_in[1];
    float* rm  = (float*)d_ws;          // 2 * B * N floats of scratch
    float* out = (float*)d_out;

    dim3 grid(Bsz * NRT);               // 1024 single-wave blocks per input
    knn_rowmean_kernel<<<grid, 32, 0, stream>>>(seed, rm);
    knn_rowmean_kernel<<<grid, 32, 0, stream>>>(gts,  rm + (size_t)Bsz * Npts);
    density_loss_reduce<<<1, 256, 0, stream>>>(rm, out);
}